// SlideWindowAttention_12996571037871
// MI455X (gfx1250) — compile-verified
//
#include <hip/hip_runtime.h>
#include <hip/hip_bf16.h>

typedef float v2f __attribute__((ext_vector_type(2)));
typedef float v8f __attribute__((ext_vector_type(8)));

__device__ __forceinline__ v8f wmma_f32(v2f a, v2f b, v8f c) {
    // V_WMMA_F32_16X16X4_F32: D = A(16x4) * B(4x16) + C(16x16), all fp32
    return __builtin_amdgcn_wmma_f32_16x16x4_f32(
        /*neg_a=*/false, a, /*neg_b=*/false, b,
        /*c_mod=*/(short)0, c, /*reuse_a=*/false, /*reuse_b=*/false);
}

// ---------------------------------------------------------------------------
// Kernel 1: fused triple projection GEMM.  grid.z selects which of
// (key@Wk, value@Wv, query@Wq) this block computes.
// Block = 256 threads (8 waves). Each wave computes a 32x64 output tile
// (2 A-fragments x 4 B-fragments = 8 accumulators), so every B fragment is
// reused across both row-halves -> ~10.7 FLOP/byte from cache.
// Block covers 256 rows x 64 cols. Grid = (N/64, M/256, 3).
// ---------------------------------------------------------------------------
__global__ void proj_gemm(const float* __restrict__ Kin,
                          const float* __restrict__ Vin,
                          const float* __restrict__ Qin,
                          const float* __restrict__ Wk,
                          const float* __restrict__ Wv,
                          const float* __restrict__ Wq,
                          float* __restrict__ kproj,
                          float* __restrict__ vproj,
                          float* __restrict__ qproj,
                          int M, int Kd, int N) {
    const int which = blockIdx.z;
    const float* __restrict__ X = (which == 0) ? Kin : (which == 1) ? Vin : Qin;
    const float* __restrict__ W = (which == 0) ? Wk  : (which == 1) ? Wv  : Wq;
    float* __restrict__ O       = (which == 0) ? kproj : (which == 1) ? vproj : qproj;

    const int lane = threadIdx.x & 31;
    const int wave = threadIdx.x >> 5;
    const int lo = lane & 15;        // N index within tile (B/C/D), M index (A)
    const int hi = lane >> 4;        // K-pair selector (A/B), row-half (C/D)

    const int tileM = (blockIdx.y * 8 + wave) * 32;   // 32 rows per wave
    const int tileN = blockIdx.x * 64;

    v8f acc00 = {}; v8f acc01 = {}; v8f acc02 = {}; v8f acc03 = {};  // rows 0..15
    v8f acc10 = {}; v8f acc11 = {}; v8f acc12 = {}; v8f acc13 = {};  // rows 16..31

    const float* arow0 = X + (size_t)(tileM + lo)      * Kd + hi * 2;
    const float* arow1 = X + (size_t)(tileM + 16 + lo) * Kd + hi * 2;

    for (int k0 = 0; k0 < Kd; k0 += 4) {
        // A fragments: lane holds X[row][k0+hi*2 .. +1]  (contiguous b64 loads)
        v2f a0 = *(const v2f*)(arow0 + k0);
        v2f a1 = *(const v2f*)(arow1 + k0);
        const float* wr0 = W + (size_t)(k0 + hi * 2) * N + tileN + lo;
        // B fragments for 4 adjacent N-tiles; loads coalesce across lanes.
        v2f b0; b0.x = wr0[ 0];      b0.y = wr0[ 0 + N];
        v2f b1; b1.x = wr0[16];      b1.y = wr0[16 + N];
        v2f b2; b2.x = wr0[32];      b2.y = wr0[32 + N];
        v2f b3; b3.x = wr0[48];      b3.y = wr0[48 + N];
        acc00 = wmma_f32(a0, b0, acc00);
        acc10 = wmma_f32(a1, b0, acc10);
        acc01 = wmma_f32(a0, b1, acc01);
        acc11 = wmma_f32(a1, b1, acc11);
        acc02 = wmma_f32(a0, b2, acc02);
        acc12 = wmma_f32(a1, b2, acc12);
        acc03 = wmma_f32(a0, b3, acc03);
        acc13 = wmma_f32(a1, b3, acc13);
    }

    // C/D layout: VGPR v -> row base + v + hi*8, lane lo -> col
    #pragma unroll
    for (int v = 0; v < 8; ++v) {
        float* orow0 = O + (size_t)(tileM + v + hi * 8)      * N + tileN + lo;
        float* orow1 = O + (size_t)(tileM + 16 + v + hi * 8) * N + tileN + lo;
        orow0[ 0] = acc00[v];
        orow0[16] = acc01[v];
        orow0[32] = acc02[v];
        orow0[48] = acc03[v];
        orow1[ 0] = acc10[v];
        orow1[16] = acc11[v];
        orow1[32] = acc12[v];
        orow1[48] = acc13[v];
    }
}

// ---------------------------------------------------------------------------
// Kernel 2: sliding-window attention over projected k/v/q.
// One 128-thread block (4 waves) per tile of 16 queries.
// The union key-window of 16 queries [t0 .. t0+15] is [t0-64, t0+79] : 144 keys.
//   Phase 1: S(16x144) = Q(16x768) * Kwin^T    (WMMA, key-tiles split on waves)
//   Phase 2: masked softmax per row, 1/L mean factor folded in (16 threads)
//   Phase 3: O(16x768) = P(16x144) * Vwin      (WMMA, col-tiles split on waves)
// ---------------------------------------------------------------------------
#define SWA_W    64
#define SWA_WLEN 144   // 129 + 15, padded union window for a 16-query tile

__global__ void swa_kernel(const float* __restrict__ kp,
                           const float* __restrict__ vp,
                           const float* __restrict__ qp,
                           float* __restrict__ out,
                           int T, int S, int D) {
    __shared__ float sc[16][SWA_WLEN];   // scores -> probabilities (9216 B)

    const int tid  = threadIdx.x;        // 128 threads = 4 waves
    const int lane = tid & 31;
    const int wave = tid >> 5;
    const int lo = lane & 15;
    const int hi = lane >> 4;

    const int t0    = blockIdx.x * 16;   // first query of this tile
    const int kbase = t0 - SWA_W;        // union window start (may be < 0)

    // ---- Phase 1: scores via WMMA ----
    const float* qrow = qp + (size_t)(t0 + lo) * D + hi * 2;  // A: M=lo
    for (int kt = wave; kt < 9; kt += 4) {                    // wave-uniform
        const int krow = kbase + kt * 16 + lo;                // B: N=lo -> key row
        const int krc  = min(max(krow, 0), S - 1);            // clamp (masked later)
        const float* krowp = kp + (size_t)krc * D + hi * 2;
        v8f acc = {};
        for (int k0 = 0; k0 < D; k0 += 4) {
            v2f a = *(const v2f*)(qrow + k0);
            v2f b = *(const v2f*)(krowp + k0);   // B[kk][n] = k[key_n][kk]
            acc = wmma_f32(a, b, acc);
        }
        #pragma unroll
        for (int v = 0; v < 8; ++v)
            sc[v + hi * 8][kt * 16 + lo] = acc[v];
    }
    __syncthreads();

    // ---- Phase 2: masked softmax + mean(1/L) fold ----
    if (tid < 16) {
        const int t = t0 + tid;
        const int start = max(0, t - SWA_W);
        const int end   = min(S, t + SWA_W + 1);
        const float scale = 0.03608439182435161f;   // 768^-0.5
        float m = -INFINITY;
        for (int c = 0; c < SWA_WLEN; ++c) {
            const int j = kbase + c;
            const float s = (j >= start && j < end) ? sc[tid][c] * scale : -INFINITY;
            sc[tid][c] = s;
            m = fmaxf(m, s);
        }
        float sum = 0.f;
        for (int c = 0; c < SWA_WLEN; ++c) {
            const float s = sc[tid][c];
            const float e = (s == -INFINITY) ? 0.f : __expf(s - m);
            sc[tid][c] = e;
            sum += e;
        }
        const float inv = 1.0f / (sum * (float)(end - start));  // softmax + mean
        for (int c = 0; c < SWA_WLEN; ++c) sc[tid][c] *= inv;
    }
    __syncthreads();

    // ---- Phase 3: O = P * Vwin via WMMA ----
    for (int ct = wave; ct < D / 16; ct += 4) {               // wave-uniform
        const int col = ct * 16 + lo;
        v8f acc = {};
        for (int k0 = 0; k0 < SWA_WLEN; k0 += 4) {
            // A[M=lo][K=k0+hi*2 .. +1] from LDS (probabilities)
            v2f a; a.x = sc[lo][k0 + hi * 2];
                   a.y = sc[lo][k0 + hi * 2 + 1];
            // B[kk][n] = v[kbase+kk][col]; clamp rows (p==0 where invalid)
            const int vr  = kbase + k0 + hi * 2;
            const int vr0 = min(max(vr,     0), S - 1);
            const int vr1 = min(max(vr + 1, 0), S - 1);
            v2f b; b.x = vp[(size_t)vr0 * D + col];
                   b.y = vp[(size_t)vr1 * D + col];
            acc = wmma_f32(a, b, acc);
        }
        #pragma unroll
        for (int v = 0; v < 8; ++v)
            out[(size_t)(t0 + v + hi * 8) * D + col] = acc[v];
    }
}

extern "C" void kernel_launch(void* const* d_in, const int* in_sizes, int n_in,
                              void* d_out, int out_size, void* d_ws, size_t ws_size,
                              hipStream_t stream) {
    const float* key   = (const float*)d_in[0];
    const float* value = (const float*)d_in[1];
    const float* query = (const float*)d_in[2];
    const float* Wk    = (const float*)d_in[3];
    const float* Wv    = (const float*)d_in[4];
    const float* Wq    = (const float*)d_in[5];
    float* out = (float*)d_out;

    const int S = 2048, T = 2048, E = 768, D = 768;

    float* ws = (float*)d_ws;
    float* kproj = ws;                       // [S, D]
    float* vproj = ws + (size_t)S * D;       // [S, D]
    float* qproj = ws + 2 * (size_t)S * D;   // [T, D]

    // Projections: grid (N/64, M/256, 3), 256 threads (8 waves) per block.
    dim3 gproj(D / 64, S / 256, 3);
    proj_gemm<<<gproj, 256, 0, stream>>>(key, value, query, Wk, Wv, Wq,
                                         kproj, vproj, qproj, S, E, D);

    // Attention: one 128-thread block per 16-query tile.
    swa_kernel<<<dim3(T / 16), 128, 0, stream>>>(kproj, vproj, qproj, out, T, S, D);
}